// DistanceWeightedSampling_55370718380617
// MI455X (gfx1250) — compile-verified
//
#include <hip/hip_runtime.h>
#include <math.h>

#define NN 4096
#define DD 512
#define BK 8
#define KM1 7
#define NT (NN * KM1)   // 28672 anchor/pos/neg triples

typedef float v2f __attribute__((ext_vector_type(2)));
typedef float v8f __attribute__((ext_vector_type(8)));

// ---- order-preserving float <-> unsigned for atomicMax on signed floats ----
__device__ __forceinline__ unsigned f2ord(float f) {
    unsigned u = __float_as_uint(f);
    return (u & 0x80000000u) ? ~u : (u | 0x80000000u);
}
__device__ __forceinline__ float ord2f(unsigned u) {
    return (u & 0x80000000u) ? __uint_as_float(u & 0x7FFFFFFFu)
                             : __uint_as_float(~u);
}

__device__ __forceinline__ unsigned long long mix64(unsigned long long z) {
    z += 0x9E3779B97F4A7C15ull;
    z = (z ^ (z >> 30)) * 0xBF58476D1CE4E5B9ull;
    z = (z ^ (z >> 27)) * 0x94D049BB133111EBull;
    return z ^ (z >> 31);
}

// ---------------- Pass 0: row squared-norms + init global max -------------
__global__ void dws_sqnorm_kernel(const float* __restrict__ x,
                                  float* __restrict__ sq,
                                  unsigned* __restrict__ gmaxu) {
    int row = blockIdx.x, tid = threadIdx.x;
    float s = 0.f;
    for (int c = tid; c < DD; c += 128) {
        float v = x[(size_t)row * DD + c];
        s += v * v;
    }
    __shared__ float sm[128];
    sm[tid] = s; __syncthreads();
    for (int k = 64; k; k >>= 1) {
        if (tid < k) sm[tid] += sm[tid + k];
        __syncthreads();
    }
    if (tid == 0) {
        sq[row] = sm[0];
        if (row == 0) *gmaxu = f2ord(-INFINITY);
    }
}

// Per-16x16-tile epilogue: distance -> log-weight, masking, store.
// Returns the (NaN-dropping) local max of the unmasked-or-not log-weights.
__device__ __forceinline__ float dws_epilogue(const v8f& acc, int ib, int j,
                                              const float* __restrict__ sq,
                                              float sqj,
                                              float* __restrict__ LW) {
    float lmax = -INFINITY;
#pragma unroll
    for (int r = 0; r < 8; ++r) {
        const int i = ib + r;
        float d2 = sq[i] + sqj - 2.0f * acc[r] + (i == j ? 1.0f : 0.0f);
        float dn = sqrtf(d2);                 // NaN if d2<0 (matches ref path)
        bool  nan_d = (dn != dn);
        float dist = fmaxf(dn, 0.5f);
        float lw = (2.0f - (float)DD) * logf(dist)
                 - 254.5f * logf(1.0f - 0.25f * dist * dist);
        lmax = fmaxf(lmax, lw);               // fmaxf drops NaN lanes
        bool masked = ((i >> 3) == (j >> 3)) || nan_d || !(dist < 1.4f);
        LW[(size_t)i * NN + j] = masked ? -INFINITY : lw;
    }
    return lmax;
}

// ---------- Pass 1: WMMA Gram, 2x4 register-blocked (32x64 per wave) ------
// Per K-step: 2 A-fragments + 4 B-fragments (6 x b64/lane) feed 8 WMMAs.
// 8 waves per block; grid covers 128 x 64 wave-regions.
__global__ void dws_gram_kernel(const float* __restrict__ x,
                                const float* __restrict__ sq,
                                float* __restrict__ LW,
                                unsigned* __restrict__ gmaxu) {
    const int lane = threadIdx.x & 31;
    const int wv   = threadIdx.x >> 5;
    const int hv   = lane >> 4;       // K half (0/1)
    const int l16  = lane & 15;       // M (for A) / N (for B)
    const int i0   = blockIdx.y * 32;                 // 2 row tiles
    const int j0   = (blockIdx.x * 8 + wv) * 64;      // 4 col tiles

    const float* ar0 = x + (size_t)(i0 + l16) * DD + 2 * hv;
    const float* ar1 = ar0 + (size_t)16 * DD;
    const float* br0 = x + (size_t)(j0 + l16) * DD + 2 * hv;
    const float* br1 = br0 + (size_t)16 * DD;
    const float* br2 = br1 + (size_t)16 * DD;
    const float* br3 = br2 + (size_t)16 * DD;

    v8f acc[8] = {};
#pragma unroll 2
    for (int k = 0; k < DD; k += 4) {
        v2f a0 = *(const v2f*)(ar0 + k);
        v2f a1 = *(const v2f*)(ar1 + k);
        v2f b0 = *(const v2f*)(br0 + k);
        v2f b1 = *(const v2f*)(br1 + k);
        v2f b2 = *(const v2f*)(br2 + k);
        v2f b3 = *(const v2f*)(br3 + k);
        acc[0] = __builtin_amdgcn_wmma_f32_16x16x4_f32(false, a0, false, b0, (short)0, acc[0], false, false);
        acc[1] = __builtin_amdgcn_wmma_f32_16x16x4_f32(false, a0, false, b1, (short)0, acc[1], false, false);
        acc[2] = __builtin_amdgcn_wmma_f32_16x16x4_f32(false, a0, false, b2, (short)0, acc[2], false, false);
        acc[3] = __builtin_amdgcn_wmma_f32_16x16x4_f32(false, a0, false, b3, (short)0, acc[3], false, false);
        acc[4] = __builtin_amdgcn_wmma_f32_16x16x4_f32(false, a1, false, b0, (short)0, acc[4], false, false);
        acc[5] = __builtin_amdgcn_wmma_f32_16x16x4_f32(false, a1, false, b1, (short)0, acc[5], false, false);
        acc[6] = __builtin_amdgcn_wmma_f32_16x16x4_f32(false, a1, false, b2, (short)0, acc[6], false, false);
        acc[7] = __builtin_amdgcn_wmma_f32_16x16x4_f32(false, a1, false, b3, (short)0, acc[7], false, false);
    }

    // Epilogue over the 2x4 tile grid
    float lmax = -INFINITY;
#pragma unroll
    for (int c = 0; c < 4; ++c) {
        const int j = j0 + 16 * c + l16;
        const float sqj = sq[j];
        lmax = fmaxf(lmax, dws_epilogue(acc[c],     i0 + 8 * hv,      j, sq, sqj, LW));
        lmax = fmaxf(lmax, dws_epilogue(acc[4 + c], i0 + 16 + 8 * hv, j, sq, sqj, LW));
    }
    // wave-level max, single atomic per wave
    for (int off = 16; off; off >>= 1)
        lmax = fmaxf(lmax, __shfl_xor(lmax, off, 32));
    if (lane == 0) atomicMax(gmaxu, f2ord(lmax));
}

// ---------- Pass 2: weights = max(exp(lw-gmax),0.01)*mask; row sums -------
__global__ void dws_weights_kernel(float* __restrict__ W,
                                   const unsigned* __restrict__ gmaxu,
                                   float* __restrict__ rowsum) {
    const int row = blockIdx.x, tid = threadIdx.x;
    const float gmax = ord2f(*gmaxu);
    float s = 0.f;
    for (int j = tid; j < NN; j += 256) {
        __builtin_prefetch(&W[(size_t)row * NN + j + 1024], 0, 0);
        float lw = W[(size_t)row * NN + j];
        float w = (lw == -INFINITY) ? 0.f : fmaxf(expf(lw - gmax), 0.01f);
        W[(size_t)row * NN + j] = w;
        s += w;
    }
    __shared__ float sm[256];
    sm[tid] = s; __syncthreads();
    for (int k = 128; k; k >>= 1) {
        if (tid < k) sm[tid] += sm[tid + k];
        __syncthreads();
    }
    if (tid == 0) rowsum[row] = sm[0];
}

// ---------- Pass 3: Gumbel-argmax categorical, 7 draws per row ------------
__global__ void dws_sample_kernel(const float* __restrict__ W,
                                  const float* __restrict__ rowsum,
                                  int* __restrict__ nidx) {
    const int t = blockIdx.x;          // t = row*7 + draw
    const int row = t / KM1;
    const int m   = t % KM1;
    const int tid = threadIdx.x;
    const bool uniform = !(rowsum[row] > 0.f);

    float best = -INFINITY;
    int   bidx = 0;
    for (int j = tid; j < NN; j += 256) {
        __builtin_prefetch(&W[(size_t)row * NN + j + 1024], 0, 0);
        float w = W[(size_t)row * NN + j];
        float logit = uniform ? 0.f : (w > 0.f ? logf(w) : -INFINITY);
        unsigned long long h =
            mix64(((unsigned long long)(m * NN + row) << 32) ^
                  (unsigned long long)j ^ 0x5EED0042A5A5ull);
        float u = (float)(h >> 40) * (1.0f / 16777216.0f)
                + (0.5f / 16777216.0f);            // (0,1)
        float g = -logf(-logf(u));
        float score = logit + g;
        if (score > best || (score == best && j < bidx)) { best = score; bidx = j; }
    }
    __shared__ float sb[256];
    __shared__ int   si[256];
    sb[tid] = best; si[tid] = bidx; __syncthreads();
    for (int k = 128; k; k >>= 1) {
        if (tid < k) {
            float ob = sb[tid + k]; int oi = si[tid + k];
            if (ob > sb[tid] || (ob == sb[tid] && oi < si[tid])) {
                sb[tid] = ob; si[tid] = oi;
            }
        }
        __syncthreads();
    }
    if (tid == 0) nidx[t] = si[0];
}

// ---------- Pass 4: write a_idx + gather x[a], x[p], x[n] ------------------
__global__ void dws_gather_kernel(const float* __restrict__ x,
                                  const int* __restrict__ nidx,
                                  float* __restrict__ out) {
    const int t = blockIdx.x;          // 0..NT-1
    const int tid = threadIdx.x;       // 128 threads, float4 copies
    const int i = t / KM1, m = t % KM1;
    const int s = i & (BK - 1);
    const int bs = i & ~(BK - 1);
    const int p = bs + m + (m >= s ? 1 : 0);
    const int nn = nidx[t];

    float* oA  = out;                                   // NT floats
    float* oXa = out + NT;                              // NT*DD
    float* oXp = oXa + (size_t)NT * DD;
    float* oXn = oXp + (size_t)NT * DD;

    if (tid == 0) oA[t] = (float)i;

    const float4* xa = (const float4*)(x + (size_t)i  * DD);
    const float4* xp = (const float4*)(x + (size_t)p  * DD);
    const float4* xn = (const float4*)(x + (size_t)nn * DD);
    float4* da = (float4*)(oXa + (size_t)t * DD);
    float4* dp = (float4*)(oXp + (size_t)t * DD);
    float4* dn = (float4*)(oXn + (size_t)t * DD);
    // DD/4 = 128 float4 per row, one per thread
    da[tid] = xa[tid];
    dp[tid] = xp[tid];
    dn[tid] = xn[tid];
}

__global__ void dws_copyx_kernel(const float* __restrict__ x,
                                 float* __restrict__ oX) {
    const int row = blockIdx.x, tid = threadIdx.x;
    const float4* src = (const float4*)(x + (size_t)row * DD);
    float4* dst = (float4*)(oX + (size_t)row * DD);
    dst[tid] = src[tid];
}

extern "C" void kernel_launch(void* const* d_in, const int* in_sizes, int n_in,
                              void* d_out, int out_size, void* d_ws, size_t ws_size,
                              hipStream_t stream) {
    (void)in_sizes; (void)n_in; (void)out_size; (void)ws_size;
    const float* x = (const float*)d_in[0];
    float* out = (float*)d_out;

    // workspace layout (float-indexed)
    float* wsf    = (float*)d_ws;
    float* LW     = wsf;                         // NN*NN (64 MB): log-weights, then weights
    float* sq     = wsf + (size_t)NN * NN;       // NN
    float* rowsum = sq + NN;                     // NN
    unsigned* gmaxu = (unsigned*)(rowsum + NN);  // 1 slot
    int* nidx     = (int*)(rowsum + NN + 16);    // NT ints

    dws_sqnorm_kernel<<<NN, 128, 0, stream>>>(x, sq, gmaxu);
    dws_gram_kernel<<<dim3(NN / (64 * 8), NN / 32), 256, 0, stream>>>(x, sq, LW, gmaxu);
    dws_weights_kernel<<<NN, 256, 0, stream>>>(LW, gmaxu, rowsum);
    dws_sample_kernel<<<NT, 256, 0, stream>>>(LW, rowsum, nidx);
    dws_gather_kernel<<<NT, 128, 0, stream>>>(x, nidx, out);
    dws_copyx_kernel<<<NN, 128, 0, stream>>>(x, out + NT + 3 * (size_t)NT * DD);
}